// Modulated_SpectralConv3d_80083960201367
// MI455X (gfx1250) — compile-verified
//
#include <hip/hip_runtime.h>

typedef __attribute__((ext_vector_type(2))) float v2f;
typedef __attribute__((ext_vector_type(8))) float v8f;
typedef __attribute__((ext_vector_type(4))) int   v4i;

#define PI_F 3.14159265358979323846f

__device__ __forceinline__ int imin(int a, int b) { return a < b ? a : b; }
__device__ __forceinline__ int imax(int a, int b) { return a > b ? a : b; }

// 16-byte global -> LDS copy: async tensor path if available, else plain copy.
__device__ __forceinline__ void lds_cp16(float* ldst, const float* gsrc) {
#if __has_builtin(__builtin_amdgcn_global_load_async_to_lds_b128)
    __builtin_amdgcn_global_load_async_to_lds_b128(
        (__attribute__((address_space(1))) v4i*)gsrc,
        (__attribute__((address_space(3))) v4i*)ldst, 0, 0);
#else
    *(float4*)ldst = *(const float4*)gsrc;
#endif
}

__device__ __forceinline__ void wait_async_lds() {
#if __has_builtin(__builtin_amdgcn_s_wait_asynccnt)
    __builtin_amdgcn_s_wait_asynccnt(0);
#elif __has_builtin(__builtin_amdgcn_global_load_async_to_lds_b128)
    asm volatile("s_wait_asynccnt 0" ::: "memory");
#endif
}

// ---------------------------------------------------------------------------
// Spectral buffers are PLANAR: float[4][32 ci][2 re/im][64 x][64 y][8 z].
// Strides (floats): b 2097152, ci 65536, comp 32768, x 512, y 8, z 1.
// ---------------------------------------------------------------------------

// Stage 1: forward real DFT along Z (64 -> bins 0..7).
// x: [4][32][64][64][64] f32. 524288 lines, 32 lines / 256-thread block.
__global__ __launch_bounds__(256)
void dft_z_fwd(const float* __restrict__ x, float* __restrict__ out) {
    __shared__ float xs[2048];
    __shared__ float tc[64], ts[64];
    int tid = threadIdx.x;
    if (tid < 64) {
        float a = (2.0f * PI_F / 64.0f) * (float)tid;
        tc[tid] = __cosf(a);
        ts[tid] = __sinf(a);
    }
    size_t base = (size_t)blockIdx.x * 2048;
    #pragma unroll
    for (int i = 0; i < 8; ++i)
        xs[i * 256 + tid] = x[base + (size_t)i * 256 + tid];
    __syncthreads();

    int line = tid >> 3;   // 0..31
    int k    = tid & 7;    // bin
    const float* p = &xs[line * 64];
    float re = 0.0f, im = 0.0f;
    #pragma unroll 8
    for (int n = 0; n < 64; ++n) {
        int idx = (k * n) & 63;
        float v = p[n];
        re += v * tc[idx];   // exp(-i 2pi k n / 64)
        im -= v * ts[idx];
    }
    int gl  = blockIdx.x * 32 + line;          // (b,c,x,y) flat
    int bc  = gl >> 12, rem = gl & 4095;
    int off = bc * 65536 + (rem >> 6) * 512 + (rem & 63) * 8;
    out[off + k]         = re;
    out[off + 32768 + k] = im;
}

// 64-point complex DFT along Y (alongY=1, stride 8) or X (stride 512), planar.
// 65536 lines, 4 lines / 256-thread block. dir = -1 fwd, +1 inverse.
__global__ __launch_bounds__(256)
void dft64_c(const float* __restrict__ in, float* __restrict__ out,
             int alongY, float dir) {
    __shared__ float2 ls[256];
    __shared__ float tc[64], ts[64];
    int tid = threadIdx.x;
    if (tid < 64) {
        float a = (2.0f * PI_F / 64.0f) * (float)tid;
        tc[tid] = __cosf(a);
        ts[tid] = dir * __sinf(a);
    }
    int lineLocal = tid >> 6;
    int j         = tid & 63;
    int l   = blockIdx.x * 4 + lineLocal;
    int bc  = l >> 9, rem = l & 511;
    int base, stride;
    if (alongY) { base = bc * 65536 + (rem >> 3) * 512 + (rem & 7); stride = 8;   }
    else        { base = bc * 65536 + rem;                          stride = 512; }

    ls[lineLocal * 64 + j] = make_float2(in[base + j * stride],
                                         in[base + j * stride + 32768]);
    __syncthreads();

    const float2* p = &ls[lineLocal * 64];
    float re = 0.0f, im = 0.0f;
    #pragma unroll 8
    for (int n = 0; n < 64; ++n) {
        int idx = (n * j) & 63;
        float2 v = p[n];
        float c = tc[idx], s = ts[idx];   // exp(i dir 2pi n j / 64)
        re += v.x * c - v.y * s;
        im += v.x * s + v.y * c;
    }
    out[base + j * stride]         = re;
    out[base + j * stride + 32768] = im;
}

__global__ __launch_bounds__(256)
void zfill(float4* __restrict__ p, int n4) {
    int i = blockIdx.x * 256 + threadIdx.x;
    if (i < n4) p[i] = make_float4(0.f, 0.f, 0.f, 0.f);
}

// ---------------------------------------------------------------------------
// Spectral conv, f32 WMMA 16x16x4, LDS-staged inputs (async-to-LDS).
// Workgroup = 128 threads (4 waves). Tile: 64 points = (xi, yc..yc+7, z 0..7).
// Wave w: oc rows [16w,16w+16) x 4 N-subtiles. Input staged in 2 phases of
// 32 channels: sIn[32 ic][5 sx][12 sy][8 z] (zero-padded halo), 60 KB LDS.
// Grid: 896 workgroups (8 regions x 4 batches x {8|48} tiles).
// ---------------------------------------------------------------------------
__global__ __launch_bounds__(128)
void spectral_conv_wmma(const float* __restrict__ Xf, float* __restrict__ Yf,
                        const float* __restrict__ w0, const float* __restrict__ w1,
                        const float* __restrict__ w2, const float* __restrict__ w3,
                        const float* __restrict__ w4, const float* __restrict__ w5,
                        const float* __restrict__ w6, const float* __restrict__ w7) {
    __shared__ float sIn[15360];     // 32*5*12*8 floats = 60 KB

    int bid = blockIdx.x;
    int r, local, tilesRB;
    if (bid < 128) { r = bid >> 5;              local = bid & 31;          tilesRB = 8;  }
    else           { int e = bid - 128; r = 4 + e / 192; local = e % 192;  tilesRB = 48; }
    int b = local / tilesRB;
    int t = local % tilesRB;

    int x0, y0, nx, ny;
    const float* w;
    switch (r) {
        case 0:  x0 = 0;  nx = 8;  y0 = 0;  ny = 8;  w = w0; break;
        case 1:  x0 = 56; nx = 8;  y0 = 0;  ny = 8;  w = w1; break;
        case 2:  x0 = 0;  nx = 8;  y0 = 56; ny = 8;  w = w2; break;
        case 3:  x0 = 56; nx = 8;  y0 = 56; ny = 8;  w = w3; break;
        case 4:  x0 = 8;  nx = 48; y0 = 0;  ny = 8;  w = w4; break;
        case 5:  x0 = 0;  nx = 8;  y0 = 8;  ny = 48; w = w5; break;
        case 6:  x0 = 8;  nx = 48; y0 = 56; ny = 8;  w = w6; break;
        default: x0 = 56; nx = 8;  y0 = 8;  ny = 48; w = w7; break;
    }
    __builtin_prefetch(w, 0, 1);

    int cpx = ny >> 3;               // y-chunks per x (1 or 6)
    int xi  = t / cpx;               // block-local x of this tile
    int yc  = (t % cpx) * 8;         // block-local y start

    int tid   = threadIdx.x;
    int lane  = tid & 31;
    int wave  = tid >> 5;
    int nIdx  = lane & 15;           // A: M row (oc); B/D: N col (point)
    int khalf = lane >> 4;           // K pair {0,1} vs {2,3}
    int zi      = nIdx & 7;          // point z
    int colBase = nIdx >> 3;         // point y-col = st*2 + colBase

    int bBase = b * 2097152;

    // Valid halo window (block-local coords).
    int lox = imax(xi - 2, 0), hix = imin(xi + 2, nx - 1);
    int loy = imax(yc - 2, 0), hiy = imin(yc + 9, ny - 1);
    int nvx = hix - lox + 1,  nvy = hiy - loy + 1;
    int nvxy = nvx * nvy;

    v8f acc[4];
    #pragma unroll
    for (int st = 0; st < 4; ++st) acc[st] = (v8f){0.f,0.f,0.f,0.f,0.f,0.f,0.f,0.f};

    for (int ph = 0; ph < 2; ++ph) {             // ph=0: re channels, ph=1: im
        __syncthreads();                         // LDS reuse fence
        #pragma unroll
        for (int i = 0; i < 30; ++i)             // zero 15360 floats as float4
            ((float4*)sIn)[i * 128 + tid] = make_float4(0.f, 0.f, 0.f, 0.f);
        __syncthreads();

        // Copy valid (ic, x, y) z-columns (32 B each) into padded LDS tile.
        int ncols = 32 * nvxy;
        for (int c = tid; c < ncols; c += 128) {
            int icl = c / nvxy;
            int rmc = c % nvxy;
            int gx  = lox + rmc / nvy;
            int gy  = loy + rmc % nvy;
            const float* g = Xf + bBase + icl * 65536 + ph * 32768
                           + (x0 + gx) * 512 + (y0 + gy) * 8;
            float* lp = sIn + ((icl * 5 + (gx - xi + 2)) * 12 + (gy - yc + 2)) * 8;
            lds_cp16(lp,     g);
            lds_cp16(lp + 4, g + 4);
        }
        wait_async_lds();
        __syncthreads();

        for (int tap = 0; tap < 125; ++tap) {
            int kx = tap / 25, ky = (tap / 5) % 5, kz = tap % 5;
            int sz = zi + kz - 2;
            bool vz = (unsigned)sz < 8u;
            int szc = vz ? sz : 0;

            int oc = wave * 16 + nIdx;
            const float* wl = w + oc * 8000 + tap;   // [oc][ic][tap], ic stride 125

            #pragma unroll 2
            for (int kc = 0; kc < 8; ++kc) {
                int icl0 = kc * 4 + khalf * 2;       // LDS-local channel
                int ic0  = ph * 32 + icl0;           // global input channel
                v2f a;
                a.x = wl[ic0 * 125];
                a.y = wl[(ic0 + 1) * 125];
                int lb0 = ((icl0 * 5 + kx) * 12 + ky) * 8 + szc;
                int lb1 = (((icl0 + 1) * 5 + kx) * 12 + ky) * 8 + szc;
                #pragma unroll
                for (int st = 0; st < 4; ++st) {
                    int coff = (st * 2 + colBase) * 8;
                    v2f bb;
                    float v0 = sIn[lb0 + coff];
                    float v1 = sIn[lb1 + coff];
                    bb.x = vz ? v0 : 0.0f;
                    bb.y = vz ? v1 : 0.0f;
                    acc[st] = __builtin_amdgcn_wmma_f32_16x16x4_f32(
                                  false, a, false, bb, (short)0, acc[st], false, false);
                }
            }
        }
    }

    // D layout: VGPR rr -> M = rr (lanes 0-15) / 8+rr (lanes 16-31); N = nIdx.
    #pragma unroll
    for (int st = 0; st < 4; ++st) {
        int col = st * 2 + colBase;
        int spa = bBase + (x0 + xi) * 512 + (y0 + yc + col) * 8 + zi;
        #pragma unroll
        for (int rr = 0; rr < 8; ++rr) {
            int oc = wave * 16 + rr + khalf * 8;
            int co = oc >> 1, cm = oc & 1;    // out chan c -> (c//2, c%2 re/im)
            Yf[spa + co * 65536 + cm * 32768] = acc[st][rr];
        }
    }
}

// Inverse real DFT along Z from 8 bins (imag(DC) ignored), scale 1/64^3.
// 524288 lines, 4 lines / 256-thread block.
__global__ __launch_bounds__(256)
void idft_z(const float* __restrict__ in, float* __restrict__ out) {
    __shared__ float2 bins[32];
    __shared__ float tc[64], ts[64];
    int tid = threadIdx.x;
    if (tid < 64) {
        float a = (2.0f * PI_F / 64.0f) * (float)tid;
        tc[tid] = __cosf(a);
        ts[tid] = __sinf(a);
    }
    size_t lbase = (size_t)blockIdx.x * 4;
    if (tid < 32) {
        int ll = tid >> 3, k = tid & 7;
        int l   = (int)lbase + ll;
        int bc  = l >> 12, rem = l & 4095;
        int off = bc * 65536 + (rem >> 6) * 512 + (rem & 63) * 8;
        bins[ll * 8 + k] = make_float2(in[off + k], in[off + 32768 + k]);
    }
    __syncthreads();

    int ll = tid >> 6;
    int n  = tid & 63;
    const float2* q = &bins[ll * 8];
    float sum = q[0].x;
    #pragma unroll
    for (int k = 1; k < 8; ++k) {
        int idx = (k * n) & 63;
        sum += 2.0f * (q[k].x * tc[idx] - q[k].y * ts[idx]);
    }
    out[(lbase + ll) * 64 + n] = sum * (1.0f / 262144.0f);
}

// ---------------------------------------------------------------------------
extern "C" void kernel_launch(void* const* d_in, const int* in_sizes, int n_in,
                              void* d_out, int out_size, void* d_ws, size_t ws_size,
                              hipStream_t stream) {
    (void)in_sizes; (void)n_in; (void)out_size; (void)ws_size;

    const float* x  = (const float*)d_in[0];
    const float* w1 = (const float*)d_in[1];
    const float* w2 = (const float*)d_in[2];
    const float* w3 = (const float*)d_in[3];
    const float* w4 = (const float*)d_in[4];
    const float* w5 = (const float*)d_in[5];
    const float* w6 = (const float*)d_in[6];
    const float* w7 = (const float*)d_in[7];
    const float* w8 = (const float*)d_in[8];

    // Two 33.5 MB planar spectral buffers: [4][32][2][64][64][8] f32.
    float* buf0 = (float*)d_ws;
    float* buf1 = buf0 + 8388608;

    // Forward: Z real-DFT (8 bins), then Y and X 64-pt complex DFTs.
    dft_z_fwd<<<16384, 256, 0, stream>>>(x, buf0);
    dft64_c<<<16384, 256, 0, stream>>>(buf0, buf1, 1, -1.0f);   // Y fwd
    dft64_c<<<16384, 256, 0, stream>>>(buf1, buf0, 0, -1.0f);   // X fwd

    // Spectral conv (octant regions) into zeroed buf1.
    zfill<<<8192, 256, 0, stream>>>((float4*)buf1, 2097152);
    spectral_conv_wmma<<<896, 128, 0, stream>>>(buf0, buf1,
                                                w1, w2, w3, w4, w5, w6, w7, w8);

    // Inverse: X, Y complex DFTs (conjugate), then real Z synthesis + scale.
    dft64_c<<<16384, 256, 0, stream>>>(buf1, buf0, 0, 1.0f);    // X inv
    dft64_c<<<16384, 256, 0, stream>>>(buf0, buf1, 1, 1.0f);    // Y inv
    idft_z<<<131072, 256, 0, stream>>>(buf1, (float*)d_out);
}